// JKNetGATConcat_36352603193548
// MI455X (gfx1250) — compile-verified
//
#include <hip/hip_runtime.h>
#include <math.h>

// ---------------------------------------------------------------------------
// JKNet-style 2-layer GAT (H=1) for MI455X / gfx1250.
// GEMMs run on the WMMA pipe in full f32 (V_WMMA_F32_16X16X4_F32).
// ---------------------------------------------------------------------------

typedef __attribute__((ext_vector_type(2))) float v2f;
typedef __attribute__((ext_vector_type(8))) float v8f;

#define DIN   128
#define CDIM  256
#define OUTSTRIDE 512
#define SLOPE 0.2f
#define ENC_NEG_INF 0x007FFFFFu   // enc(-inf)

__device__ __forceinline__ unsigned enc_f32(float f) {
    unsigned b = __float_as_uint(f);
    return (b & 0x80000000u) ? ~b : (b | 0x80000000u);
}
__device__ __forceinline__ float dec_f32(unsigned u) {
    unsigned b = (u & 0x80000000u) ? (u & 0x7FFFFFFFu) : ~u;
    return __uint_as_float(b);
}

// --------------------------- init ------------------------------------------
__global__ void init_kernel(float* __restrict__ agg, float* __restrict__ denom,
                            unsigned* __restrict__ m_enc, int Nn, int totNC) {
    int i = blockIdx.x * blockDim.x + threadIdx.x;
    if (i < totNC) agg[i] = 0.0f;
    if (i < Nn) { denom[i] = 0.0f; m_enc[i] = ENC_NEG_INF; }
}

// --------------------------- WMMA GEMM (NT, f32) ---------------------------
// Hout[m, n] = sum_k A[m, k] * B[n, k]        A: [Nrows, K] (row stride lda)
//                                             B: [Cout , K] (row stride K)
// One wave computes one 16x16 tile; K stepped by 4 via V_WMMA_F32_16X16X4_F32.
__global__ void gemm_nt_wmma(const float* __restrict__ A, int lda,
                             const float* __restrict__ B,
                             float* __restrict__ Hout,
                             int Nrows, int K, int Cout) {
    int wave = (int)((blockIdx.x * blockDim.x + threadIdx.x) >> 5);
    int lane = threadIdx.x & 31;
    int tilesN   = Cout >> 4;
    int numTiles = (Nrows >> 4) * tilesN;
    if (wave >= numTiles) return;              // wave-uniform: EXEC stays all-1s

    int tm = wave / tilesN;
    int tn = wave % tilesN;
    int m0 = tm << 4, n0 = tn << 4;

    int r    = lane & 15;                      // row within 16
    int koff = (lane >> 4) << 1;               // lanes 0-15 -> K=0,1 ; 16-31 -> K=2,3

    const float* arow = A + (size_t)(m0 + r) * lda + koff;
    const float* brow = B + (size_t)(n0 + r) * K   + koff;

    v8f acc = {};
    for (int k = 0; k < K; k += 4) {
        v2f af = *(const v2f*)(arow + k);
        v2f bf = *(const v2f*)(brow + k);
        acc = __builtin_amdgcn_wmma_f32_16x16x4_f32(
            /*neg_a=*/false, af, /*neg_b=*/false, bf,
            /*c_mod=*/(short)0, acc, /*reuse_a=*/false, /*reuse_b=*/false);
    }

    // C/D layout: lanes 0-15 -> M=j, N=lane ; lanes 16-31 -> M=8+j, N=lane-16
    int rowBase = m0 + ((lane >> 4) << 3);
    int col     = n0 + r;
    float* outp = Hout + (size_t)rowBase * Cout + col;
#pragma unroll
    for (int j = 0; j < 8; ++j)
        outp[(size_t)j * Cout] = acc[j];
}

// --------------------- per-node attention dot products ---------------------
__global__ void node_dots(const float* __restrict__ h,
                          const float* __restrict__ a_src,
                          const float* __restrict__ a_dst,
                          float* __restrict__ e_src, float* __restrict__ e_dst,
                          int Nn) {
    int wave = (int)((blockIdx.x * blockDim.x + threadIdx.x) >> 5);
    int lane = threadIdx.x & 31;
    if (wave >= Nn) return;
    const float* hr = h + (size_t)wave * CDIM;
    float ss = 0.0f, sd = 0.0f;
#pragma unroll
    for (int j = 0; j < CDIM / 32; ++j) {
        int c = lane + 32 * j;
        float v = hr[c];
        ss += v * a_src[c];
        sd += v * a_dst[c];
    }
    for (int off = 16; off > 0; off >>= 1) {
        ss += __shfl_down(ss, off, 32);
        sd += __shfl_down(sd, off, 32);
    }
    if (lane == 0) { e_src[wave] = ss; e_dst[wave] = sd; }
}

// --------------------- edge pass 1: segment max ----------------------------
__global__ void edge_max(const int* __restrict__ ei, int E_, int Nn,
                         const float* __restrict__ e_src,
                         const float* __restrict__ e_dst,
                         unsigned* __restrict__ m_enc) {
    int i = blockIdx.x * blockDim.x + threadIdx.x;
    int Etot = E_ + Nn;
    if (i >= Etot) return;
    int s = (i < E_) ? ei[i]      : (i - E_);
    int d = (i < E_) ? ei[E_ + i] : (i - E_);
    float x = e_src[s] + e_dst[d];
    x = (x > 0.0f) ? x : SLOPE * x;
    atomicMax(m_enc + d, enc_f32(x));
}

// --------------------- edge pass 2: exp + segment sum ----------------------
__global__ void edge_exp(const int* __restrict__ ei, int E_, int Nn,
                         const float* __restrict__ e_src,
                         const float* __restrict__ e_dst,
                         const unsigned* __restrict__ m_enc,
                         float* __restrict__ p, float* __restrict__ denom) {
    int i = blockIdx.x * blockDim.x + threadIdx.x;
    int Etot = E_ + Nn;
    if (i >= Etot) return;
    int s = (i < E_) ? ei[i]      : (i - E_);
    int d = (i < E_) ? ei[E_ + i] : (i - E_);
    float x = e_src[s] + e_dst[d];
    x = (x > 0.0f) ? x : SLOPE * x;
    float pv = expf(x - dec_f32(m_enc[d]));
    p[i] = pv;
    atomicAdd(denom + d, pv);
}

// --------------------- edge pass 3: weighted aggregation -------------------
// One wave per edge; lane strides over the 256 channels (8 per lane).
__global__ void edge_aggregate(const int* __restrict__ ei, int E_, int Nn,
                               const float* __restrict__ p,
                               const float* __restrict__ denom,
                               const float* __restrict__ h,
                               float* __restrict__ agg) {
    int wave = (int)((blockIdx.x * blockDim.x + threadIdx.x) >> 5);
    int lane = threadIdx.x & 31;
    int Etot = E_ + Nn;
    if (wave >= Etot) return;
    int s = (wave < E_) ? ei[wave]      : (wave - E_);
    int d = (wave < E_) ? ei[E_ + wave] : (wave - E_);
    float alpha = p[wave] / (denom[d] + 1e-16f);
    const float* hs = h + (size_t)s * CDIM;
    float* ag = agg + (size_t)d * CDIM;
#pragma unroll
    for (int j = 0; j < CDIM / 32; ++j) {
        int c = lane + 32 * j;
        atomicAdd(ag + c, hs[c] * alpha);
    }
}

// --------------------- finalize: bias + ReLU -> d_out slice ----------------
__global__ void finalize(const float* __restrict__ agg,
                         const float* __restrict__ bias,
                         float* __restrict__ out, int colOff, int totNC) {
    int i = blockIdx.x * blockDim.x + threadIdx.x;
    if (i >= totNC) return;
    int n = i >> 8;           // / CDIM
    int c = i & (CDIM - 1);
    float v = agg[i] + bias[c];
    v = (v > 0.0f) ? v : 0.0f;
    out[(size_t)n * OUTSTRIDE + colOff + c] = v;
}

// ---------------------------------------------------------------------------
extern "C" void kernel_launch(void* const* d_in, const int* in_sizes, int n_in,
                              void* d_out, int out_size, void* d_ws, size_t ws_size,
                              hipStream_t stream) {
    const float* x      = (const float*)d_in[0];
    const int*   ei     = (const int*)  d_in[1];
    const float* W1     = (const float*)d_in[2];
    const float* a1_src = (const float*)d_in[3];
    const float* a1_dst = (const float*)d_in[4];
    const float* b1     = (const float*)d_in[5];
    const float* W2     = (const float*)d_in[6];
    const float* a2_src = (const float*)d_in[7];
    const float* a2_dst = (const float*)d_in[8];
    const float* b2     = (const float*)d_in[9];
    float* out = (float*)d_out;

    const int Nn   = in_sizes[0] / DIN;   // 50000
    const int E_   = in_sizes[1] / 2;     // 500000
    const int Etot = E_ + Nn;
    const int totNC = Nn * CDIM;

    // -------- workspace carve-up (256B aligned slabs) ----------------------
    char* ws = (char*)d_ws;
    size_t off = 0;
    auto alloc = [&](size_t bytes) -> char* {
        char* pp = ws + off;
        off += (bytes + 255) & ~(size_t)255;
        return pp;
    };
    float*    h     = (float*)   alloc((size_t)totNC * 4);
    float*    agg   = (float*)   alloc((size_t)totNC * 4);
    float*    es    = (float*)   alloc((size_t)Nn * 4);
    float*    ed    = (float*)   alloc((size_t)Nn * 4);
    unsigned* m_enc = (unsigned*)alloc((size_t)Nn * 4);
    float*    denom = (float*)   alloc((size_t)Nn * 4);
    float*    p     = (float*)   alloc((size_t)Etot * 4);
    (void)ws_size;

    const int TPB = 256;
    const int gInit  = (totNC + TPB - 1) / TPB;
    const int gTileW = ((Nn >> 4) * (CDIM >> 4) + 7) / 8;   // 8 waves / block
    const int gNodeW = (Nn + 7) / 8;
    const int gEdge  = (Etot + TPB - 1) / TPB;
    const int gEdgeW = (Etot + 7) / 8;

    // ============================ layer 1 ==================================
    init_kernel<<<gInit, TPB, 0, stream>>>(agg, denom, m_enc, Nn, totNC);
    gemm_nt_wmma<<<gTileW, TPB, 0, stream>>>(x, DIN, W1, h, Nn, DIN, CDIM);
    node_dots<<<gNodeW, TPB, 0, stream>>>(h, a1_src, a1_dst, es, ed, Nn);
    edge_max<<<gEdge, TPB, 0, stream>>>(ei, E_, Nn, es, ed, m_enc);
    edge_exp<<<gEdge, TPB, 0, stream>>>(ei, E_, Nn, es, ed, m_enc, p, denom);
    edge_aggregate<<<gEdgeW, TPB, 0, stream>>>(ei, E_, Nn, p, denom, h, agg);
    finalize<<<gInit, TPB, 0, stream>>>(agg, b1, out, /*colOff=*/0, totNC);

    // ============================ layer 2 ==================================
    // x1 lives in d_out cols [0,256) with row stride 512.
    init_kernel<<<gInit, TPB, 0, stream>>>(agg, denom, m_enc, Nn, totNC);
    gemm_nt_wmma<<<gTileW, TPB, 0, stream>>>(out, OUTSTRIDE, W2, h, Nn, CDIM, CDIM);
    node_dots<<<gNodeW, TPB, 0, stream>>>(h, a2_src, a2_dst, es, ed, Nn);
    edge_max<<<gEdge, TPB, 0, stream>>>(ei, E_, Nn, es, ed, m_enc);
    edge_exp<<<gEdge, TPB, 0, stream>>>(ei, E_, Nn, es, ed, m_enc, p, denom);
    edge_aggregate<<<gEdgeW, TPB, 0, stream>>>(ei, E_, Nn, p, denom, h, agg);
    finalize<<<gInit, TPB, 0, stream>>>(agg, b2, out, /*colOff=*/CDIM, totNC);
}